// EdgePreprocess_75685913690197
// MI455X (gfx1250) — compile-verified
//
#include <hip/hip_runtime.h>

typedef __attribute__((ext_vector_type(2))) float v2f;
typedef __attribute__((ext_vector_type(8))) float v8f;

// -------- workspace layout (floats) --------
// [0,            nbatch*9)        sym_strain S (row-major per batch)
// [nbatch*9,     2*nbatch*9)      cell_eff = cell + cell @ S
// [2*nbatch*9,   ... + 3*natoms)  pos_new  = pos + pos @ S[batch]

__device__ inline float sel3(float a0, float a1, float a2, int idx) {
    return idx == 0 ? a0 : (idx == 1 ? a1 : (idx == 2 ? a2 : 0.0f));
}

// One wave. For each batch b: S = 0.5*(strain+strain^T); cell_eff = cell + cell@S.
// The 3x3x3 product is computed with V_WMMA_F32_16X16X4_F32 (zero-padded tiles).
__global__ __launch_bounds__(32) void cell_strain_kernel(
    const float* __restrict__ cell, const float* __restrict__ strain,
    float* __restrict__ ws_S, float* __restrict__ ws_cell, int nbatch)
{
    const int lane = threadIdx.x;      // 0..31, full wave => EXEC all ones at WMMA
    const int mn   = lane & 15;        // row (A) / col (B) index
    const int half = lane >> 4;        // K half: 0 -> K={0,1}, 1 -> K={2,3}

    for (int b = 0; b < nbatch; ++b) {
        const float* st = strain + b * 9;
        const float* cl = cell   + b * 9;

        // sym strain, fully unrolled in registers
        float S00 = st[0];
        float S01 = 0.5f * (st[1] + st[3]);
        float S02 = 0.5f * (st[2] + st[6]);
        float S11 = st[4];
        float S12 = 0.5f * (st[5] + st[7]);
        float S22 = st[8];
        // S10=S01, S20=S02, S21=S12

        // ---- A operand: 16x4 f32, A[m][k]=cell[m][k] for m,k<3 else 0 ----
        // lane layout (ISA 7.12.2): VGPR0 -> K=2*half, VGPR1 -> K=2*half+1, row=mn
        const int r = (mn < 3) ? mn : 0;                 // clamped, in-range address
        float ca0 = cl[r * 3 + 2 * half];                // K = 0 or 2 (always < 3)
        float ca1 = cl[r * 3 + 1];                       // K = 1 (only valid if half==0)
        v2f a;
        a.x = (mn < 3) ? ca0 : 0.0f;
        a.y = (mn < 3 && half == 0) ? ca1 : 0.0f;

        // ---- B operand: 4x16 f32, B[k][n]=S[k][n] for k,n<3 else 0 ----
        // VGPR0 -> K=2*half, VGPR1 -> K=2*half+1, col=mn
        float b0 = (half == 0) ? sel3(S00, S01, S02, mn)   // K=0 row
                               : sel3(S02, S12, S22, mn);  // K=2 row
        float b1 = (half == 0) ? sel3(S01, S11, S12, mn)   // K=1 row
                               : 0.0f;                     // K=3 row (pad)
        v2f bb; bb.x = b0; bb.y = b1;

        v8f c = {};
        c = __builtin_amdgcn_wmma_f32_16x16x4_f32(
                /*neg_a=*/false, a, /*neg_b=*/false, bb,
                /*c_mod=*/(short)0, c, /*reuse_a=*/false, /*reuse_b=*/false);

        // D layout: VGPR r holds row r (lanes 0-15, N=lane). Need rows/cols 0..2.
        if (lane < 3) {
            ws_cell[b * 9 + 0 * 3 + lane] = cl[0 * 3 + lane] + c[0];
            ws_cell[b * 9 + 1 * 3 + lane] = cl[1 * 3 + lane] + c[1];
            ws_cell[b * 9 + 2 * 3 + lane] = cl[2 * 3 + lane] + c[2];
        }
        // store full sym strain (9 values) for the atom kernel
        if (lane < 9) {
            int i = lane / 3, j = lane % 3;
            ws_S[b * 9 + lane] = 0.5f * (st[i * 3 + j] + st[j * 3 + i]);
        }
    }
}

// pos_new[n] = pos[n] + pos[n] @ S[batch[n]]
__global__ __launch_bounds__(256) void pos_strain_kernel(
    const float* __restrict__ pos, const int* __restrict__ batch,
    const float* __restrict__ ws_S, float* __restrict__ pos_new, int natoms)
{
    int n = blockIdx.x * blockDim.x + threadIdx.x;
    if (n >= natoms) return;
    int b = batch[n];
    const float* S = ws_S + b * 9;     // tiny, L2-resident
    float px = pos[3 * n + 0], py = pos[3 * n + 1], pz = pos[3 * n + 2];
    // out_j = sum_i p_i * S[i][j]
    float dx = px * S[0] + py * S[3] + pz * S[6];
    float dy = px * S[1] + py * S[4] + pz * S[7];
    float dz = px * S[2] + py * S[5] + pz * S[8];
    pos_new[3 * n + 0] = px + dx;
    pos_new[3 * n + 1] = py + dy;
    pos_new[3 * n + 2] = pz + dz;
}

// Per edge: edge_vec = pos_new[dst]-pos_new[src] + shift @ cell_eff[batch[src]]
// Streams (shift, indices, outputs) use non-temporal hints so the L2 keeps the
// 6MB pos_new gather table resident; cell_eff lives in LDS.
__global__ __launch_bounds__(256) void edge_kernel(
    const float* __restrict__ pos_new, const float* __restrict__ cell_eff,
    const float* __restrict__ cell_shift, const int* __restrict__ eidx,
    const int* __restrict__ batch, float* __restrict__ out,
    long long E, int nbatch)
{
    __shared__ float sC[128 * 9];
    const int ncf = nbatch * 9;
    for (int i = threadIdx.x; i < ncf; i += blockDim.x) sC[i] = cell_eff[i];
    __syncthreads();

    long long e = (long long)blockIdx.x * blockDim.x + threadIdx.x;
    if (e >= E) return;

    // prefetch streaming data well ahead (speculative RT prefetch)
    long long ep = e + 32768; if (ep >= E) ep = E - 1;
    __builtin_prefetch(&cell_shift[3 * ep], 0, 1);
    __builtin_prefetch(&eidx[ep], 0, 1);
    __builtin_prefetch(&eidx[E + ep], 0, 1);

    int src = __builtin_nontemporal_load(&eidx[e]);
    int dst = __builtin_nontemporal_load(&eidx[E + e]);
    float sx = __builtin_nontemporal_load(&cell_shift[3 * e + 0]);
    float sy = __builtin_nontemporal_load(&cell_shift[3 * e + 1]);
    float sz = __builtin_nontemporal_load(&cell_shift[3 * e + 2]);

    int b = batch[src];                       // L2-resident gather
    const float* C = sC + b * 9;              // LDS

    float psx = pos_new[3 * (long long)src + 0];
    float psy = pos_new[3 * (long long)src + 1];
    float psz = pos_new[3 * (long long)src + 2];
    float pdx = pos_new[3 * (long long)dst + 0];
    float pdy = pos_new[3 * (long long)dst + 1];
    float pdz = pos_new[3 * (long long)dst + 2];

    // v_j = sum_i shift_i * cell_eff[i][j]
    float vx = sx * C[0] + sy * C[3] + sz * C[6];
    float vy = sx * C[1] + sy * C[4] + sz * C[7];
    float vz = sx * C[2] + sy * C[5] + sz * C[8];

    float evx = pdx - psx + vx;
    float evy = pdy - psy + vy;
    float evz = pdz - psz + vz;

    __builtin_nontemporal_store(evx, &out[3 * e + 0]);
    __builtin_nontemporal_store(evy, &out[3 * e + 1]);
    __builtin_nontemporal_store(evz, &out[3 * e + 2]);
    __builtin_nontemporal_store(sqrtf(evx * evx + evy * evy + evz * evz),
                                &out[3 * E + e]);
}

extern "C" void kernel_launch(void* const* d_in, const int* in_sizes, int n_in,
                              void* d_out, int out_size, void* d_ws, size_t ws_size,
                              hipStream_t stream) {
    const float* pos        = (const float*)d_in[0];
    const float* cell       = (const float*)d_in[1];
    const float* cell_shift = (const float*)d_in[2];
    const int*   edge_index = (const int*)d_in[3];
    const int*   batch      = (const int*)d_in[4];
    const float* strain     = (const float*)d_in[5];

    const int       natoms = in_sizes[0] / 3;
    const int       nbatch = in_sizes[1] / 9;
    const long long E      = (long long)(in_sizes[2] / 3);

    float* ws      = (float*)d_ws;
    float* ws_S    = ws;                       // nbatch*9
    float* ws_cell = ws + (size_t)nbatch * 9;  // nbatch*9
    float* ws_pos  = ws + (size_t)2 * nbatch * 9;

    cell_strain_kernel<<<1, 32, 0, stream>>>(cell, strain, ws_S, ws_cell, nbatch);

    pos_strain_kernel<<<(natoms + 255) / 256, 256, 0, stream>>>(
        pos, batch, ws_S, ws_pos, natoms);

    edge_kernel<<<(int)((E + 255) / 256), 256, 0, stream>>>(
        ws_pos, ws_cell, cell_shift, edge_index, batch,
        (float*)d_out, E, nbatch);
}